// CurvatureAwareGCN_90056874262563
// MI455X (gfx1250) — compile-verified
//
#include <hip/hip_runtime.h>
#include <math.h>

#define NN 100000
#define EE 1600000
#define DD 64

typedef __attribute__((ext_vector_type(2))) float v2f;
typedef __attribute__((ext_vector_type(8))) float v8f;

// ---------------------------------------------------------------------------
// Kernel 0: zero the workspace (summed[N*64] + cnt[N]) every launch.
// ---------------------------------------------------------------------------
__global__ void zero_ws_kernel(float* __restrict__ ws, int n) {
    int i = blockIdx.x * blockDim.x + threadIdx.x;
    if (i < n) ws[i] = 0.0f;
}

// ---------------------------------------------------------------------------
// Kernel 1: per-edge gate + gather + scatter-add.
// 64 threads per edge (one per feature dim), 4 edges per 256-thread block.
// x, summed, curvature are all L2-resident on MI455X (192 MB L2), so the
// gathers and the f32 atomics resolve in L2; only edge_index streams from HBM.
// ---------------------------------------------------------------------------
__global__ __launch_bounds__(256) void edge_scatter_kernel(
    const float* __restrict__ x, const float* __restrict__ curv,
    const float* __restrict__ Wc, const float* __restrict__ bc,
    const long long* __restrict__ rowp, const long long* __restrict__ colp,
    float* __restrict__ summed, float* __restrict__ cnt) {
    const int group = threadIdx.x >> 6;          // 0..3 : edge within block
    const int d     = threadIdx.x & 63;          // feature dim
    const long long e = (long long)blockIdx.x * 4 + group;

    // Prefetch the edge-index stream ahead (global_prefetch_b8).
    if (threadIdx.x == 0 && e + 8192 < EE) {
        __builtin_prefetch(&rowp[e + 8192], 0, 0);
        __builtin_prefetch(&colp[e + 8192], 0, 0);
    }

    const int r = (int)rowp[e];
    const int c = (int)colp[e];
    const float cd   = fabsf(curv[r] - curv[c]);
    const float z    = cd * Wc[d] + bc[d];
    const float gate = 1.0f / (1.0f + __expf(-z));
    const float msg  = x[(long long)c * DD + d] * gate;

    unsafeAtomicAdd(&summed[(long long)r * DD + d], msg);   // global_atomic_add_f32
    if (d == 0) unsafeAtomicAdd(&cnt[r], 1.0f);
}

// ---------------------------------------------------------------------------
// Kernel 2: out = gelu(mean @ W_lin^T + b_lin) via V_WMMA_F32_16X16X4_F32.
// One wave per 16x16 output tile; 4 waves (one block) cover one 16-row strip.
// mean = summed * (1/max(cnt,1)) is fused into the A-fragment load.
// ---------------------------------------------------------------------------
__global__ __launch_bounds__(128) void mean_gemm_gelu_kernel(
    const float* __restrict__ summed, const float* __restrict__ cnt,
    const float* __restrict__ W, const float* __restrict__ bl,
    float* __restrict__ out) {
    const int lane = threadIdx.x & 31;
    const int wave = threadIdx.x >> 5;           // n-tile 0..3
    const int m0 = blockIdx.x * 16;
    const int n0 = wave * 16;
    const int hr   = lane & 15;                  // row-in-tile (A) / col-in-tile (B)
    const int koff = (lane >= 16) ? 2 : 0;       // half-wave K split for f32 frags

    // Per-lane constants across the K loop.
    const int arow = m0 + hr;                    // mean row this lane feeds into A
    const float inv = 1.0f / fmaxf(cnt[arow], 1.0f);
    const float* __restrict__ srow = summed + (long long)arow * DD;
    const float* __restrict__ wrow = W + (long long)(n0 + hr) * DD;  // B[k][n]=W[n][k]

    v8f acc = {};
#pragma unroll
    for (int k0 = 0; k0 < DD; k0 += 4) {
        v2f a, b;
        a.x = srow[k0 + koff]     * inv;
        a.y = srow[k0 + koff + 1] * inv;
        b.x = wrow[k0 + koff];
        b.y = wrow[k0 + koff + 1];
        // D = A(16x4,f32) x B(4x16,f32) + C(16x16,f32)
        acc = __builtin_amdgcn_wmma_f32_16x16x4_f32(
            /*neg_a=*/false, a, /*neg_b=*/false, b,
            /*c_mod=*/(short)0, acc, /*reuse_a=*/false, /*reuse_b=*/false);
    }

    // Epilogue: acc[j] = D[M = j + (lane<16?0:8)][N = hr]; fuse bias + exact GELU.
    const int n = n0 + hr;
    const float bias = bl[n];
    const int mbase = m0 + ((lane < 16) ? 0 : 8);
#pragma unroll
    for (int j = 0; j < 8; ++j) {
        float v = acc[j] + bias;
        float g = 0.5f * v * (1.0f + erff(v * 0.70710678118654752f));
        out[(long long)(mbase + j) * DD + n] = g;
    }
}

// ---------------------------------------------------------------------------
extern "C" void kernel_launch(void* const* d_in, const int* in_sizes, int n_in,
                              void* d_out, int out_size, void* d_ws, size_t ws_size,
                              hipStream_t stream) {
    const float*     x    = (const float*)d_in[0];
    const float*     curv = (const float*)d_in[1];
    const float*     Wc   = (const float*)d_in[2];   // [64,1]
    const float*     bc   = (const float*)d_in[3];   // [64]
    const float*     Wl   = (const float*)d_in[4];   // [64,64]
    const float*     bl   = (const float*)d_in[5];   // [64]
    const long long* ei   = (const long long*)d_in[6];  // [2,E] int64
    float* out = (float*)d_out;

    float* summed = (float*)d_ws;                 // N*64 f32
    float* cnt    = summed + (size_t)NN * DD;     // N f32

    const int ztotal = NN * DD + NN;
    zero_ws_kernel<<<(ztotal + 255) / 256, 256, 0, stream>>>(summed, ztotal);

    edge_scatter_kernel<<<EE / 4, 256, 0, stream>>>(
        x, curv, Wc, bc, ei, ei + EE, summed, cnt);

    mean_gemm_gelu_kernel<<<NN / 16, 128, 0, stream>>>(summed, cnt, Wl, bl, out);
}